// STEContinuousReconstructor_67516885893765
// MI455X (gfx1250) — compile-verified
//
#include <hip/hip_runtime.h>
#include <hip/hip_bf16.h>

typedef __attribute__((ext_vector_type(2))) float v2f;
typedef __attribute__((ext_vector_type(8))) float v8f;

#define T_LEN   16384
#define B_LEN   64
#define CHUNK   256
#define HALO    32
#define NSRC    (CHUNK + HALO)   // 288 sources incl. causal halo
#define D_LEN   30
#define DPAD    32               // columns 30,31 zero-filled -> branch-free WMMA gathers
#define PSTRIDE 33               // gcd(33,64)==1 -> LDS bank-conflict free

// Native trans ops: v_log_f32 (base-2) and v_exp_f32 (base-2)
__device__ __forceinline__ float fast_log2(float v) { return __builtin_amdgcn_logf(v); }
__device__ __forceinline__ float fast_exp2(float v) { return __builtin_amdgcn_exp2f(v); }

// LOG2T[d] = log2(d + 0.5)
__device__ __constant__ float LOG2T[D_LEN] = {
    -1.00000000f, 0.58496250f, 1.32192809f, 1.80735492f, 2.16992500f,
     2.45943162f, 2.70043972f, 2.90689060f, 3.08746284f, 3.24792751f,
     3.39231742f, 3.52356196f, 3.64385619f, 3.75488750f, 3.85798100f,
     3.95419631f, 4.04439412f, 4.12928302f, 4.20945337f, 4.28540222f,
     4.35755200f, 4.42626475f, 4.49185310f, 4.55458885f, 4.61470984f,
     4.67242534f, 4.72792045f, 4.78135971f, 4.83289001f, 4.88264305f
};

__global__ __launch_bounds__(256)
void ste_recon_kernel(const float* __restrict__ x,
                      const float* __restrict__ beta_mean,
                      const float* __restrict__ beta_prec,
                      float* __restrict__ out)
{
    __shared__ float P[NSRC * PSTRIDE];   // normalized*scaled bells, [src][d]
    __shared__ float outBuf[CHUNK];

    const int tid  = threadIdx.x;
    const int c0   = blockIdx.x * CHUNK;      // first output index of this chunk
    const int bRow = blockIdx.y;              // batch row
    const float* __restrict__ xb = x + (size_t)bRow * 3 * T_LEN;

    const float bm0 = beta_mean[0], bm1 = beta_mean[1];
    const float bp0 = beta_prec[0], bp1 = beta_prec[1];

    // ---------------- Phase 1: per-source bell profiles into LDS -------------
    for (int s = tid; s < NSRC; s += 256) {
        const int t = c0 + s - HALO;          // global source position (< T always)
        float w = 0.f, dur = 0.f;
        if (t >= 0) {
            const float m = xb[t];
            bool kill = false;
            if (t > 0)         kill = kill || (m <= xb[t - 1]);
            if (t < T_LEN - 1) kill = kill || (m <  xb[t + 1]);
            const bool peak = (!kill) && (m >= 0.5f);
            const float d0v = xb[2 * T_LEN + t];
            dur = (d0v == 0.f) ? 0.f : fminf(fmaxf(d0v, 4.0f), 30.0f);
            w   = peak ? xb[T_LEN + t] : 0.f;
        }

        const float L     = (dur > 0.f) ? fast_log2(dur) : 0.f;
        const float means = bm0 + bm1 * dur;
        const float prec  = bp0 + bp1 * dur;
        const float am1   = means * prec - 1.0f;           // a - 1
        const float bm1e  = (1.0f - means) * prec - 1.0f;  // b - 1

        float bells[D_LEN];
        float ssum = 0.f;
        #pragma unroll
        for (int d = 0; d < D_LEN; ++d) {
            const float td = (float)d + 0.5f;
            float bell = 0.f;
            if (td < dur) {   // nt in (0,1); nt>=1 or dur==0 -> bell 0
                const float e = am1 * (LOG2T[d] - L)
                              + bm1e * (fast_log2(dur - td) - L);
                bell = fast_exp2(e);
            }
            bells[d] = bell;
            ssum += bell;
        }
        const float scale = (ssum != 0.f) ? (w / ssum) : 0.f;
        #pragma unroll
        for (int d = 0; d < D_LEN; ++d)
            P[s * PSTRIDE + d] = bells[d] * scale;
        // zero pad columns 30,31 so phase-2 K=30/31 gathers are branch-free
        P[s * PSTRIDE + 30] = 0.f;
        P[s * PSTRIDE + 31] = 0.f;
    }
    __syncthreads();

    // -------- Phase 2: 30-wide diagonal band-sum via V_WMMA_F32_16X16X4_F32 --
    // recon[j] = sum_d P[j + HALO - d][d].  Build A(16x4) = shifted-diagonal
    // tile columns, B = all-ones -> D[m,n] = sum_k A[m,k]; 8 accumulating
    // WMMAs cover d=0..31 (columns 30,31 are zero in LDS).
    const int lane  = tid & 31;
    const int wave  = tid >> 5;
    const int m     = lane & 15;               // A-matrix row (M)
    const int khalf = (lane >> 4) << 1;        // lanes 16-31 hold K=2,3

    v2f ones; ones.x = 1.0f; ones.y = 1.0f;

    #pragma unroll
    for (int tt = 0; tt < 2; ++tt) {           // 8 waves x 2 tiles = 256 outputs
        const int j0 = (wave * 2 + tt) * 16;
        v8f acc = {0.f, 0.f, 0.f, 0.f, 0.f, 0.f, 0.f, 0.f};
        #pragma unroll
        for (int c = 0; c < 8; ++c) {
            const int d0 = 4 * c + khalf;
            const int d1 = d0 + 1;
            v2f a;
            a.x = P[(j0 + m + HALO - d0) * PSTRIDE + d0];
            a.y = P[(j0 + m + HALO - d1) * PSTRIDE + d1];
            acc = __builtin_amdgcn_wmma_f32_16x16x4_f32(
                false, a, false, ones, (short)0, acc, false, false);
        }
        // D layout: VGPR r -> M=r (lanes 0-15, N=lane) / M=8+r (lanes 16-31).
        // Lanes with N==0 carry all 16 row sums between them.
        if ((lane & 15) == 0) {
            const int mbase = (lane >> 4) * 8;
            #pragma unroll
            for (int r = 0; r < 8; ++r)
                outBuf[j0 + mbase + r] = acc[r];
        }
    }
    __syncthreads();

    // ---------------- Coalesced stores of (recon, noisy=recon) ---------------
    const size_t base = (size_t)bRow * T_LEN + (size_t)c0 + tid;
    const float v = outBuf[tid];
    out[base] = v;
    out[(size_t)B_LEN * T_LEN + base] = v;
}

extern "C" void kernel_launch(void* const* d_in, const int* in_sizes, int n_in,
                              void* d_out, int out_size, void* d_ws, size_t ws_size,
                              hipStream_t stream) {
    (void)in_sizes; (void)n_in; (void)d_ws; (void)ws_size; (void)out_size;
    const float* x         = (const float*)d_in[0];
    const float* beta_mean = (const float*)d_in[1];
    const float* beta_prec = (const float*)d_in[2];
    float* out = (float*)d_out;

    dim3 grid(T_LEN / CHUNK, B_LEN);   // 64 x 64 workgroups
    dim3 block(256);                   // 8 wave32s
    ste_recon_kernel<<<grid, block, 0, stream>>>(x, beta_mean, beta_prec, out);
}